// GINConvolution_61538291417221
// MI455X (gfx1250) — compile-verified
//
#include <hip/hip_runtime.h>

typedef __attribute__((ext_vector_type(2))) float v2f;
typedef __attribute__((ext_vector_type(4))) float v4f;
typedef __attribute__((ext_vector_type(8))) float v8f;

#define D_IN  128
#define UNITS 128

// ---------------------------------------------------------------------------
// Kernel 1: h = x @ W  (WMMA f32 16x16x4), and seed acc = (1+eps)*h + bias.
// Block = 256 threads = 8 waves. Block computes rows [32*b, 32*b+32) x 128 cols.
// Wave w owns the 16-column slab [16w, 16w+16) and carries TWO 16x16
// accumulators (rows 0-15 and 16-31 of the tile) that share each B fragment:
// 2x the WMMA work per W load, two independent accumulation chains for ILP.
// ---------------------------------------------------------------------------
__global__ __launch_bounds__(256) void gin_gemm_kernel(
    const float* __restrict__ x, const float* __restrict__ W,
    const float* __restrict__ bias, const float* __restrict__ eps,
    float* __restrict__ h, float* __restrict__ acc, int n_nodes)
{
    // 32 rows x 128 cols of x, padded to 132 floats/row (528B = 33*16: rows stay
    // 16B-aligned, and the 132-float lane stride avoids LDS bank conflicts).
    __shared__ float xs[32][132];

    const int tid  = threadIdx.x;
    const int wave = tid >> 5;
    const int lane = tid & 31;
    const int row0 = blockIdx.x * 32;

    // Cooperative coalesced stage of the 32x128 x-tile into LDS (float4).
    const bool full_tile = (row0 + 32 <= n_nodes);
    for (int i = tid; i < 32 * 32; i += 256) {
        const int r  = i >> 5;      // row within tile
        const int c4 = i & 31;      // float4 column
        const int gr = row0 + r;
        v4f v = {0.f, 0.f, 0.f, 0.f};
        if (gr < n_nodes)
            v = *(const v4f*)(x + (size_t)gr * D_IN + (size_t)c4 * 4);
        *(v4f*)(&xs[r][c4 * 4]) = v;
    }
    __syncthreads();

    // Fragment coordinates per ISA layout tables (wave32).
    const int m    = lane & 15;           // A: row M for this lane
    const int koff = (lane >> 4) * 2;     // A/B: lanes 16-31 hold K+2,K+3
    const int n    = lane & 15;           // B/D: column N for this lane
    const int col0 = wave * 16;

    const float* Wp = W + (size_t)col0 + n;   // walk down column (col0+n) of W

    v8f c0 = {};
    v8f c1 = {};
#pragma unroll 4
    for (int k = 0; k < D_IN; k += 4) {
        // A fragments: 16x4 f32 -> 2 VGPRs/lane (ds_load_b64, 8B aligned).
        v2f a0 = *(const v2f*)(&xs[m][k + koff]);
        v2f a1 = *(const v2f*)(&xs[m + 16][k + koff]);
        // B fragment: 4x16 f32 -> 2 VGPRs/lane, row K striped across lanes.
        v2f b;
        b.x = Wp[(size_t)(k + koff)     * UNITS];
        b.y = Wp[(size_t)(k + koff + 1) * UNITS];
        c0 = __builtin_amdgcn_wmma_f32_16x16x4_f32(
                 false, a0, false, b, (short)0, c0, false, false);
        c1 = __builtin_amdgcn_wmma_f32_16x16x4_f32(
                 false, a1, false, b, (short)0, c1, false, false);
    }

    // Epilogue: D layout — VGPR v: lanes 0-15 hold M=v, lanes 16-31 hold M=v+8.
    const float scale = 1.0f + eps[0];
    const int   mbase = (lane >> 4) * 8;
    const float bv    = bias[col0 + n];

    const size_t base = (size_t)(row0 + mbase) * UNITS + (size_t)(col0 + n);
    float* __restrict__ hp = h   + base;
    float* __restrict__ ap = acc + base;

    if (full_tile) {
        // Branchless: single base pointer + compile-time immediate offsets
        // (v*512B and (v+16)*512B), so stores clause up cleanly.
#pragma unroll
        for (int v = 0; v < 8; ++v) {
            const float h0 = c0[v];
            const float h1 = c1[v];
            hp[(size_t)v * UNITS]        = h0;
            hp[(size_t)(v + 16) * UNITS] = h1;
            ap[(size_t)v * UNITS]        = scale * h0 + bv;
            ap[(size_t)(v + 16) * UNITS] = scale * h1 + bv;
        }
    } else {
        // Ragged tail (at most one block when n_nodes % 32 != 0).
#pragma unroll
        for (int v = 0; v < 8; ++v) {
            if (row0 + mbase + v < n_nodes) {
                const float h0 = c0[v];
                hp[(size_t)v * UNITS] = h0;
                ap[(size_t)v * UNITS] = scale * h0 + bv;
            }
            if (row0 + mbase + v + 16 < n_nodes) {
                const float h1 = c1[v];
                hp[(size_t)(v + 16) * UNITS] = h1;
                ap[(size_t)(v + 16) * UNITS] = scale * h1 + bv;
            }
        }
    }
}

// ---------------------------------------------------------------------------
// Kernel 2: COO scatter  acc[row[e]] += vals[e] * h[col[e]]
// One wave per edge; lane l owns features [4l, 4l+4): b128 gather + 4 f32
// atomics. row/col/val are wave-uniform -> force into SGPRs so the gather is
// saddr-based and the atomics share a scalar base. h and acc are both
// L2-resident (51.2 MB each vs 192 MB L2), so this stays on-chip.
// ---------------------------------------------------------------------------
__global__ __launch_bounds__(256) void gin_edge_kernel(
    const float* __restrict__ h, const float* __restrict__ vals,
    const int* __restrict__ row, const int* __restrict__ col,
    float* __restrict__ acc, int n_edges)
{
    const int e = blockIdx.x * 8 + (threadIdx.x >> 5);
    if (e >= n_edges) return;
    const int lane = threadIdx.x & 31;

    const int r = __builtin_amdgcn_readfirstlane(row[e]);
    const int s = __builtin_amdgcn_readfirstlane(col[e]);
    const float v =
        __int_as_float(__builtin_amdgcn_readfirstlane(__float_as_int(vals[e])));

    const v4f hv = *(const v4f*)(h + (size_t)s * UNITS + (size_t)lane * 4);
    float* op = acc + (size_t)r * UNITS + (size_t)lane * 4;

    __hip_atomic_fetch_add(op + 0, v * hv.x, __ATOMIC_RELAXED, __HIP_MEMORY_SCOPE_AGENT);
    __hip_atomic_fetch_add(op + 1, v * hv.y, __ATOMIC_RELAXED, __HIP_MEMORY_SCOPE_AGENT);
    __hip_atomic_fetch_add(op + 2, v * hv.z, __ATOMIC_RELAXED, __HIP_MEMORY_SCOPE_AGENT);
    __hip_atomic_fetch_add(op + 3, v * hv.w, __ATOMIC_RELAXED, __HIP_MEMORY_SCOPE_AGENT);
}

// ---------------------------------------------------------------------------
// Kernel 3: in-place ReLU on the accumulator (vectorized b128).
// ---------------------------------------------------------------------------
__global__ __launch_bounds__(256) void gin_relu_kernel(float* __restrict__ out, long n4)
{
    const long i = (long)blockIdx.x * blockDim.x + threadIdx.x;
    if (i < n4) {
        v4f* p = (v4f*)out;
        v4f v = p[i];
        v.x = fmaxf(v.x, 0.f);
        v.y = fmaxf(v.y, 0.f);
        v.z = fmaxf(v.z, 0.f);
        v.w = fmaxf(v.w, 0.f);
        p[i] = v;
    }
}

extern "C" void kernel_launch(void* const* d_in, const int* in_sizes, int n_in,
                              void* d_out, int out_size, void* d_ws, size_t ws_size,
                              hipStream_t stream)
{
    const float* x      = (const float*)d_in[0];
    const float* kernel = (const float*)d_in[1];
    const float* bias   = (const float*)d_in[2];
    const float* eps    = (const float*)d_in[3];
    const float* vals   = (const float*)d_in[4];
    const int*   row    = (const int*)d_in[5];
    const int*   col    = (const int*)d_in[6];
    float*       out    = (float*)d_out;

    const int n_nodes = in_sizes[0] / D_IN;
    const int n_edges = in_sizes[4];

    float* h = (float*)d_ws;   // n_nodes*UNITS floats of scratch for h

    // 1) h = x@W, acc = (1+eps)*h + bias
    const int gemm_blocks = (n_nodes + 31) / 32;
    gin_gemm_kernel<<<gemm_blocks, 256, 0, stream>>>(x, kernel, bias, eps, h, out, n_nodes);

    // 2) acc[row] += vals * h[col]   (8 edges per 256-thread block)
    const int edge_blocks = (n_edges + 7) / 8;
    gin_edge_kernel<<<edge_blocks, 256, 0, stream>>>(h, vals, row, col, out, n_edges);

    // 3) out = relu(acc)
    const long n4 = (long)out_size / 4;
    const int relu_blocks = (int)((n4 + 255) / 256);
    gin_relu_kernel<<<relu_blocks, 256, 0, stream>>>(out, n4);
}